// Matrixfactorization_59279138619492
// MI455X (gfx1250) — compile-verified
//
#include <hip/hip_runtime.h>
#include <hip/hip_bf16.h>

// Matrix-factorization batched dot product, CDNA5 (gfx1250) version.
//
// Strategy (memory-bound, AI ~0.25 FLOP/B):
//  - One wave32 processes a tile of 16 samples.
//  - 16 user rows + 16 item rows (128 f32 each) staged to LDS with fully
//    coalesced global_load_b128 (scalar row base via readfirstlane).
//  - Dot products computed as D = U(16x128) x I^T(128x16) with 32 chained
//    V_WMMA_F32_16X16X4_F32 ops; diagonal D[b][b] is the answer.
//  - LDS row stride of 132 floats makes the WMMA fragment reads
//    (16 rows @ same K, +2 K-offset for upper half-wave) bank-conflict free.

typedef float v2f __attribute__((ext_vector_type(2)));
typedef float v4f __attribute__((ext_vector_type(4)));
typedef float v8f __attribute__((ext_vector_type(8)));

#define MF_DIM       128
#define MF_LDSTRIDE  132           // 128 + 4 pad: rows start on distinct banks
#define MF_MU        3.53f
#define MF_WAVES_PER_BLOCK 2

__global__ __launch_bounds__(64) void
mf_dot_wmma_kernel(const int* __restrict__ users,
                   const int* __restrict__ items,
                   const float* __restrict__ ulat,
                   const float* __restrict__ ilat,
                   const float* __restrict__ ubias,
                   const float* __restrict__ ibias,
                   float* __restrict__ out,
                   int batch, int n_tiles)
{
    // Per-wave staging: [wave][matrix(u/i)][16 rows][132 floats] = 16896 B/wave
    __shared__ float stage[MF_WAVES_PER_BLOCK * 2 * 16 * MF_LDSTRIDE];

    const int lane    = threadIdx.x & 31;
    const int waveBlk = threadIdx.x >> 5;
    const int gwave   = blockIdx.x * MF_WAVES_PER_BLOCK + waveBlk;
    const int nwaves  = gridDim.x * MF_WAVES_PER_BLOCK;

    float* su = &stage[waveBlk * (2 * 16 * MF_LDSTRIDE)];
    float* si = su + 16 * MF_LDSTRIDE;

    const int rm    = lane & 15;        // row (A) / col (B) this lane serves
    const int half2 = (lane >> 4) * 2;  // K sub-offset: 0 for lanes 0-15, 2 for 16-31

    for (int tile = gwave; tile < n_tiles; tile += nwaves) {
        const int base = tile * 16;

        // ---- Stage 16 user rows + 16 item rows into LDS (coalesced) ----
        #pragma unroll
        for (int r = 0; r < 16; ++r) {
            int sidx = base + r;
            int cl   = sidx < batch ? sidx : batch - 1;     // clamp tail
            cl = __builtin_amdgcn_readfirstlane(cl);        // force SGPR
            int urow = users[cl] - 1;                       // scalar loads
            int irow = items[cl] - 1;
            const float* up = ulat + (size_t)urow * MF_DIM;
            const float* ip = ilat + (size_t)irow * MF_DIM;
            v4f uu = *(const v4f*)(up + lane * 4);          // global_load_b128
            v4f vv = *(const v4f*)(ip + lane * 4);
            *(v4f*)(su + r * MF_LDSTRIDE + lane * 4) = uu;  // ds_store_b128
            *(v4f*)(si + r * MF_LDSTRIDE + lane * 4) = vv;
        }

        // Wave-internal ordering of LDS stores vs. fragment loads.
        __builtin_amdgcn_fence(__ATOMIC_RELEASE, "wavefront");
        __builtin_amdgcn_wave_barrier();
        __builtin_amdgcn_fence(__ATOMIC_ACQUIRE, "wavefront");

        // ---- D(16x16) = U(16x128) x I^T(128x16), K in chunks of 4 ----
        // A layout (16x4 f32): lane L holds A[L&15][(L>>4)*2 + {0,1}]
        // B layout (4x16 f32): lane L holds B[(L>>4)*2 + {0,1}][L&15]
        //   with B[k][n] = item_row_n[k0+k]  -> same LDS addressing as A.
        const float* pa = su + rm * MF_LDSTRIDE + half2;
        const float* pb = si + rm * MF_LDSTRIDE + half2;

        v8f acc = {};
        #pragma unroll
        for (int c = 0; c < MF_DIM / 4; ++c) {
            v2f a = *(const v2f*)(pa + 4 * c);   // ds_load_b64, conflict-free
            v2f b = *(const v2f*)(pb + 4 * c);
            acc = __builtin_amdgcn_wmma_f32_16x16x4_f32(
                /*neg_a=*/false, a, /*neg_b=*/false, b,
                /*c_mod=*/(short)0, acc,
                /*reuse_a=*/false, /*reuse_b=*/false);
        }

        // ---- Extract diagonal D[s][s] ----
        // C/D layout: vgpr r, lanes 0-15 -> (M=r,   N=lane),
        //                     lanes 16-31 -> (M=r+8, N=lane-16).
        // Sample s: s<8  -> lane s,     vgpr s
        //           s>=8 -> lane s+16,  vgpr s-8
        // Owner lanes are therefore 0..7 and 24..31; vgpr index = lane<16 ? lane : lane-24.
        int dsel = (lane < 16) ? lane : (lane - 24);
        float diag = acc[0];
        #pragma unroll
        for (int r = 1; r < 8; ++r) diag = (dsel == r) ? acc[r] : diag;

        bool owner = (lane < 8) || (lane >= 24);
        int  s     = lane & 15;
        int  sidx  = base + s;
        if (owner && sidx < batch) {
            int urow = users[sidx] - 1;
            int irow = items[sidx] - 1;
            out[sidx] = diag + ubias[urow] + ibias[irow] + MF_MU;
        }
    }
}

extern "C" void kernel_launch(void* const* d_in, const int* in_sizes, int n_in,
                              void* d_out, int out_size, void* d_ws, size_t ws_size,
                              hipStream_t stream) {
    const int*   users = (const int*)  d_in[0];
    const int*   items = (const int*)  d_in[1];
    const float* ulat  = (const float*)d_in[2];
    const float* ilat  = (const float*)d_in[3];
    const float* ub    = (const float*)d_in[4];
    const float* ib    = (const float*)d_in[5];
    float*       out   = (float*)d_out;

    const int batch   = in_sizes[0];              // 1,000,000
    const int n_tiles = (batch + 15) / 16;        // 62,500

    const int blocks  = 2048;                     // 4096 waves total
    dim3 grid(blocks), block(64);                 // 2 waves / block, 33 KB LDS
    hipLaunchKernelGGL(mf_dot_wmma_kernel, grid, block, 0, stream,
                       users, items, ulat, ilat, ub, ib, out, batch, n_tiles);
}